// HGIN_25786983645584
// MI455X (gfx1250) — compile-verified
//
#include <hip/hip_runtime.h>
#include <stdint.h>

// ---------------------------------------------------------------------------
// HGIN forward on MI455X (gfx1250, wave32).
//   agg[d] += nfeats[s] once per unique (s,d)   -> hash dedup + atomicAdd (L2)
//   h = (1+alpha)*nfeats + agg                  -> fused into MLP kernel
//   out = relu(h@W1 + b1) @ W2 + b2             -> V_WMMA_F32_16X16X4_F32
// D_IN = D_OUT = 128, N_NODES = 50000 (= 3125 * 16), E = 800000.
// ---------------------------------------------------------------------------

#define D 128
#define HASH_CAP (1u << 21)   // 2M slots for <=800K keys, load factor ~0.38

typedef float v2f __attribute__((ext_vector_type(2)));
typedef float v8f __attribute__((ext_vector_type(8)));

// ---------------------------------------------------------------- zero scratch
__global__ void hgin_zero_ws(uint32_t* __restrict__ ws, size_t nwords) {
    size_t i      = (size_t)blockIdx.x * blockDim.x + threadIdx.x;
    size_t stride = (size_t)gridDim.x * blockDim.x;
    for (; i < nwords; i += stride) ws[i] = 0u;
}

// ------------------------------------------------------------------ edge dedup
// One thread per edge. key = src*N + dst + 1 (0 = empty slot sentinel).
// First inserter of a key sets flags[e]=1; duplicates get 0.
__global__ void hgin_dedup(const int* __restrict__ src,
                           const int* __restrict__ dst,
                           int nedges, int nnodes,
                           unsigned long long* __restrict__ table,
                           unsigned char* __restrict__ flags) {
    int e = blockIdx.x * blockDim.x + threadIdx.x;
    if (e >= nedges) return;
    unsigned long long key =
        (unsigned long long)(unsigned)src[e] * (unsigned long long)(unsigned)nnodes +
        (unsigned long long)(unsigned)dst[e] + 1ull;
    // splitmix64-style mixer
    unsigned long long h = key;
    h ^= h >> 33; h *= 0xff51afd7ed558ccdull;
    h ^= h >> 33; h *= 0xc4ceb9fe1a85ec53ull;
    h ^= h >> 33;
    unsigned idx = (unsigned)h & (HASH_CAP - 1u);
    unsigned char f = 0;
    for (;;) {
        unsigned long long old = atomicCAS(&table[idx], 0ull, key);
        if (old == 0ull) { f = 1; break; }   // first occurrence of this key
        if (old == key)  { f = 0; break; }   // duplicate
        idx = (idx + 1u) & (HASH_CAP - 1u);  // linear probe
    }
    flags[e] = f;
}

// -------------------------------------------------------------- edge scatter
// One wave (32 lanes) per edge; each lane moves 4 contiguous floats (float4
// load, 4x atomicAdd into agg -- agg fits in the 192MB L2, atomics stay there).
__global__ void hgin_scatter(const float* __restrict__ nfeats,
                             const int* __restrict__ src,
                             const int* __restrict__ dst,
                             const unsigned char* __restrict__ flags,
                             float* __restrict__ agg, int nedges) {
    int t = blockIdx.x * blockDim.x + threadIdx.x;
    int e = t >> 5;
    if (e >= nedges) return;
    if (!flags[e]) return;
    int lane = t & 31;
    int s = src[e], d = dst[e];
    const float4 v = *(const float4*)(nfeats + (size_t)s * D + lane * 4);
    float* a = agg + (size_t)d * D + lane * 4;
    atomicAdd(a + 0, v.x);
    atomicAdd(a + 1, v.y);
    atomicAdd(a + 2, v.z);
    atomicAdd(a + 3, v.w);
}

// ------------------------------------------------------------------ fused MLP
// 256 threads = 8 waves. Workgroup owns a 16-row tile; wave w owns output
// columns [16w, 16w+16). Per layer: 32x V_WMMA_F32_16X16X4_F32 over K=128.
//
// Fragment layouts (ISA 7.12.2, f32):
//   A 16x4:  lanes 0-15  -> row M=lane,    VGPR0=K0, VGPR1=K1
//            lanes 16-31 -> row M=lane-16, VGPR0=K2, VGPR1=K3
//   B 4x16:  lanes 0-15  -> col N=lane,    VGPR0=K0, VGPR1=K1
//            lanes 16-31 -> col N=lane-16, VGPR0=K2, VGPR1=K3
//   C/D 16x16: VGPR j -> M = j + 8*(lane>=16), N = lane&15
__global__ __launch_bounds__(256) void hgin_mlp(
    const float* __restrict__ nfeats, const float* __restrict__ agg,
    const float* __restrict__ W1, const float* __restrict__ b1,
    const float* __restrict__ W2, const float* __restrict__ b2,
    const float* __restrict__ alphaPtr, float* __restrict__ out, int nrows) {
    __shared__ float h0[16 * D];   // layer-0 activations (8 KB)
    __shared__ float h1[16 * D];   // post-ReLU layer-1 activations (8 KB)

    const int   r0  = blockIdx.x * 16;
    const float opa = 1.0f + alphaPtr[0];

    // Stage h0 = (1+alpha)*nfeats + agg  (512 float4 / 256 threads = 2 each)
    for (int i = threadIdx.x; i < 16 * (D / 4); i += 256) {
        int row = i / (D / 4);
        int c4  = (i % (D / 4)) * 4;
        float4 h = {0.f, 0.f, 0.f, 0.f};
        if (r0 + row < nrows) {
            size_t g = (size_t)(r0 + row) * D + c4;
            float4 nf = *(const float4*)(nfeats + g);
            float4 ag = *(const float4*)(agg + g);
            h.x = opa * nf.x + ag.x;
            h.y = opa * nf.y + ag.y;
            h.z = opa * nf.z + ag.z;
            h.w = opa * nf.w + ag.w;
        }
        *(float4*)(&h0[row * D + c4]) = h;
    }
    __syncthreads();

    const int wave = threadIdx.x >> 5;           // 0..7 -> N-tile
    const int lane = threadIdx.x & 31;
    const int col  = (wave << 4) + (lane & 15);  // output column (0..127)
    const int koff = (lane >> 4) << 1;           // K offset: 0 (lo half) / 2 (hi)
    const int mrow = lane & 15;                  // A-fragment row
    const int mbas = (lane >> 4) << 3;           // C/D row base: 0 or 8

    // ---- Layer 1: relu(h0 @ W1 + b1) -> h1 (LDS) ----
    v8f acc = {0.f, 0.f, 0.f, 0.f, 0.f, 0.f, 0.f, 0.f};
#pragma unroll 4
    for (int k = 0; k < D; k += 4) {
        v2f a, b;
        a.x = h0[mrow * D + k + koff];
        a.y = h0[mrow * D + k + koff + 1];
        b.x = W1[(size_t)(k + koff) * D + col];
        b.y = W1[(size_t)(k + koff + 1) * D + col];
        acc = __builtin_amdgcn_wmma_f32_16x16x4_f32(
            false, a, false, b, (short)0, acc, false, false);
    }
    {
        const float bias = b1[col];
#pragma unroll
        for (int j = 0; j < 8; ++j) {
            float v = acc[j] + bias;
            h1[(j + mbas) * D + col] = v > 0.f ? v : 0.f;
        }
    }
    __syncthreads();

    // ---- Layer 2: h1 @ W2 + b2 -> out ----
    v8f acc2 = {0.f, 0.f, 0.f, 0.f, 0.f, 0.f, 0.f, 0.f};
#pragma unroll 4
    for (int k = 0; k < D; k += 4) {
        v2f a, b;
        a.x = h1[mrow * D + k + koff];
        a.y = h1[mrow * D + k + koff + 1];
        b.x = W2[(size_t)(k + koff) * D + col];
        b.y = W2[(size_t)(k + koff + 1) * D + col];
        acc2 = __builtin_amdgcn_wmma_f32_16x16x4_f32(
            false, a, false, b, (short)0, acc2, false, false);
    }
    {
        const float bias = b2[col];
#pragma unroll
        for (int j = 0; j < 8; ++j) {
            int row = r0 + j + mbas;
            if (row < nrows)
                out[(size_t)row * D + col] = acc2[j] + bias;
        }
    }
}

// ---------------------------------------------------------------------------
extern "C" void kernel_launch(void* const* d_in, const int* in_sizes, int n_in,
                              void* d_out, int out_size, void* d_ws, size_t ws_size,
                              hipStream_t stream) {
    const float* nfeats = (const float*)d_in[0];
    const int*   src    = (const int*)d_in[1];
    const int*   dst    = (const int*)d_in[2];
    const float* W1     = (const float*)d_in[3];
    const float* b1     = (const float*)d_in[4];
    const float* W2     = (const float*)d_in[5];
    const float* b2     = (const float*)d_in[6];
    const float* alpha  = (const float*)d_in[7];
    float*       out    = (float*)d_out;

    const int nnodes = in_sizes[0] / D;   // 50000
    const int nedges = in_sizes[1];       // 800000

    // Workspace layout: [ agg : N*D f32 ][ hash : HASH_CAP u64 ][ flags : E u8 ]
    const size_t agg_bytes  = (size_t)nnodes * D * sizeof(float);
    const size_t hash_bytes = (size_t)HASH_CAP * sizeof(unsigned long long);
    float*              agg   = (float*)d_ws;
    unsigned long long* table = (unsigned long long*)((char*)d_ws + agg_bytes);
    unsigned char*      flags = (unsigned char*)((char*)d_ws + agg_bytes + hash_bytes);

    // 1) zero agg + hash table (flags are fully overwritten by dedup)
    size_t zero_words = (agg_bytes + hash_bytes) / 4;
    hgin_zero_ws<<<2048, 256, 0, stream>>>((uint32_t*)d_ws, zero_words);

    // 2) dedup: one thread per edge
    hgin_dedup<<<(nedges + 255) / 256, 256, 0, stream>>>(
        src, dst, nedges, nnodes, table, flags);

    // 3) scatter: one wave32 per edge
    long long sc_threads = (long long)nedges * 32;
    hgin_scatter<<<(int)((sc_threads + 255) / 256), 256, 0, stream>>>(
        nfeats, src, dst, flags, agg, nedges);

    // 4) fused MLP with f32 WMMA: one workgroup per 16 rows
    hgin_mlp<<<(nnodes + 15) / 16, 256, 0, stream>>>(
        nfeats, agg, W1, b1, W2, b2, alpha, out, nnodes);
}